// SimplifiedGNN_66958540145066
// MI455X (gfx1250) — compile-verified
//
#include <hip/hip_runtime.h>
#include <hip/hip_bf16.h>
#include <stdint.h>

#define D_FEAT 64
#define TILE   256          // edges staged per block (4 KB of int64 indices)

// ---------------------------------------------------------------------------
// CDNA5 async global->LDS copy (16B per lane) + ASYNCcnt wait, via inline asm
// (portable across ROCm-7.2 / amdgpu-toolchain builtin-arity differences).
// LDS operand: low 32 bits of the shared-aperture generic address = LDS offset
// ---------------------------------------------------------------------------
__device__ __forceinline__ uint32_t lds_off32(const void* p) {
    return (uint32_t)(uintptr_t)p;
}
__device__ __forceinline__ void async_copy16_to_lds(uint32_t lds_off,
                                                    const void* gptr) {
    uint64_t ga = (uint64_t)(uintptr_t)gptr;
    asm volatile("global_load_async_to_lds_b128 %0, %1, off"
                 :: "v"(lds_off), "v"(ga)
                 : "memory");
}
__device__ __forceinline__ void wait_asynccnt0() {
    asm volatile("s_wait_asynccnt 0x0" ::: "memory");
}

// ---------------------------------------------------------------------------
// Kernel 1: degree histogram over targets (col). u32 atomics, L2-resident.
// ---------------------------------------------------------------------------
__global__ void gnn_deg_kernel(const long long* __restrict__ col,
                               unsigned int* __restrict__ deg, int E) {
    int e = blockIdx.x * blockDim.x + threadIdx.x;
    if (e < E) {
        if (e + 4096 < E) __builtin_prefetch(&col[e + 4096], 0, 0);
        atomicAdd(&deg[(int)col[e]], 1u);
    }
}

// ---------------------------------------------------------------------------
// Kernel 2: dis[i] = deg>0 ? rsqrt(deg) : 0   (deg integer >=1 => max redundant)
// ---------------------------------------------------------------------------
__global__ void gnn_dis_kernel(const unsigned int* __restrict__ deg,
                               float* __restrict__ dis, int N) {
    int i = blockIdx.x * blockDim.x + threadIdx.x;
    if (i < N) {
        unsigned int d = deg[i];
        dis[i] = (d > 0u) ? __frsqrt_rn((float)d) : 0.0f;
    }
}

// ---------------------------------------------------------------------------
// Kernel 3: out = res_scale * x  (fused residual; float4-vectorized)
// ---------------------------------------------------------------------------
__global__ void gnn_init_out_kernel(const float* __restrict__ x,
                                    const float* __restrict__ res_scale_p,
                                    float* __restrict__ out, int n4) {
    int i = blockIdx.x * blockDim.x + threadIdx.x;
    if (i < n4) {
        float s = *res_scale_p;
        float4 v = reinterpret_cast<const float4*>(x)[i];
        v.x *= s; v.y *= s; v.z *= s; v.w *= s;
        reinterpret_cast<float4*>(out)[i] = v;
    }
}

// ---------------------------------------------------------------------------
// Kernel 4: edge scatter. Each block stages TILE edges of row/col into LDS via
// async-DMA (256 lanes x 16B = 4KB), waits ASYNCcnt, then 16 lanes per edge:
//   w = alpha * dis[row] * dis[col];  out[col,:] += w * x[row,:]
// Gather = coalesced 256B global_load_b128 burst; scatter = non-returning
// global_atomic_add_f32 resolving in MI455X's 192MB L2 (whole set resident).
// ---------------------------------------------------------------------------
__global__ void __launch_bounds__(256)
gnn_edge_kernel(const long long* __restrict__ row,
                const long long* __restrict__ col,
                const float* __restrict__ x,
                const float* __restrict__ dis,
                const float* __restrict__ alpha_p,
                float* __restrict__ out, int E) {
    __shared__ long long lrow[TILE];
    __shared__ long long lcol[TILE];

    const int tid      = threadIdx.x;
    const int tileBase = blockIdx.x * TILE;

    // --- stage edge indices: threads 0..127 -> row halves, 128..255 -> col ---
    {
        const int  t    = tid & 127;            // 16B slot within the half
        long long  gidx = (long long)tileBase + 2 * t;
        if (gidx > (long long)E - 2) gidx = (long long)E - 2;  // clamp tail
        const long long* src = (tid < 128) ? row  : col;
        long long*       dst = (tid < 128) ? lrow : lcol;
        async_copy16_to_lds(lds_off32(dst + 2 * t), src + gidx);
    }
    wait_asynccnt0();
    __syncthreads();

    const float alpha = *alpha_p;
    const int   eg = tid >> 4;   // edge slot within 16-edge wavefront group
    const int   f4 = tid & 15;   // which float4 of the 64-feature row

#pragma unroll 4
    for (int it = 0; it < TILE / 16; ++it) {
        const int el = it * 16 + eg;
        const int e  = tileBase + el;
        if (e < E) {
            const int r = (int)lrow[el];     // ds_load (L0-latency)
            const int c = (int)lcol[el];
            const float w = alpha * dis[r] * dis[c];

            const float4 xv =
                reinterpret_cast<const float4*>(x + (size_t)r * D_FEAT)[f4];

            float* o = out + (size_t)c * D_FEAT + (size_t)f4 * 4;
            atomicAdd(o + 0, w * xv.x);
            atomicAdd(o + 1, w * xv.y);
            atomicAdd(o + 2, w * xv.z);
            atomicAdd(o + 3, w * xv.w);
        }
    }
}

// ---------------------------------------------------------------------------
// Launcher. Inputs (dict order): x[f32 N*64], alpha[f32 1], res_scale[f32 1],
// edge_index[int64 2*E] (row = first E, col = second E). Output: f32 N*64.
// Scratch in d_ws: [0,4N) deg (u32), [4N,8N) dis (f32).
// ---------------------------------------------------------------------------
extern "C" void kernel_launch(void* const* d_in, const int* in_sizes, int n_in,
                              void* d_out, int out_size, void* d_ws, size_t ws_size,
                              hipStream_t stream) {
    const float*     x       = (const float*)d_in[0];
    const float*     alpha_p = (const float*)d_in[1];
    const float*     res_p   = (const float*)d_in[2];
    const long long* edges   = (const long long*)d_in[3];

    const int N = in_sizes[0] / D_FEAT;      // 100000
    const int E = in_sizes[3] / 2;           // 1600000
    const long long* row = edges;            // sources
    const long long* col = edges + E;        // targets

    unsigned int* deg = (unsigned int*)d_ws;
    float*        dis = (float*)((char*)d_ws + (size_t)N * sizeof(unsigned int));

    // zero degree counters (graph-capture-safe async memset)
    hipMemsetAsync(deg, 0, (size_t)N * sizeof(unsigned int), stream);

    const int B = 256; // 8 wave32 waves per block

    // 1) degree histogram
    gnn_deg_kernel<<<(E + B - 1) / B, B, 0, stream>>>(col, deg, E);

    // 2) deg^{-1/2}
    gnn_dis_kernel<<<(N + B - 1) / B, B, 0, stream>>>(deg, dis, N);

    // 3) out = res_scale * x  (fused residual)
    const int n4 = N * (D_FEAT / 4);
    gnn_init_out_kernel<<<(n4 + B - 1) / B, B, 0, stream>>>(x, res_p,
                                                            (float*)d_out, n4);

    // 4) async-staged edge scatter-add
    const int nblocks = (E + TILE - 1) / TILE;   // 6250 blocks
    gnn_edge_kernel<<<nblocks, B, 0, stream>>>(row, col, x, dis, alpha_p,
                                               (float*)d_out, E);
}